// PointNet_SA_module_basic_4389456577470
// MI455X (gfx1250) — compile-verified
//
#include <hip/hip_runtime.h>

typedef __attribute__((ext_vector_type(2))) float v2f;
typedef __attribute__((ext_vector_type(8))) float v8f;

#define B_      16
#define N_      8192
#define NPOINT_ 1024
#define NSAMPLE_ 32
#define FEAT_   64
#define RADIUS2 0.04f
#define BIGF    1.0e10f

// ---------------------------------------------------------------------------
// Kernel 1: farthest point sampling. One block per batch, 1024 threads.
// Each thread keeps its 8 points + running min-distance in registers.
// Per iteration: distance update + wave32 shfl argmax + cross-wave LDS argmax.
// ---------------------------------------------------------------------------
__global__ __launch_bounds__(1024) void fps_kernel(const float* __restrict__ xyz,
                                                   int* __restrict__ cents) {
  const int b = blockIdx.x;
  const int tid = threadIdx.x;
  const float* bx = xyz + (size_t)b * N_ * 3;

  float px[8], py[8], pz[8], dist[8];
#pragma unroll
  for (int k = 0; k < 8; ++k) {
    const int p = tid + k * 1024;
    px[k] = bx[p * 3 + 0];
    py[k] = bx[p * 3 + 1];
    pz[k] = bx[p * 3 + 2];
    dist[k] = BIGF;
  }

  __shared__ int s_far;
  __shared__ float s_cx, s_cy, s_cz;
  __shared__ float s_val[32];
  __shared__ int s_idx[32];
  if (tid == 0) s_far = 0;
  __syncthreads();

  for (int it = 0; it < NPOINT_; ++it) {
    if (tid == 0) {
      const int f = s_far;
      cents[b * NPOINT_ + it] = f;
      s_cx = bx[f * 3 + 0];
      s_cy = bx[f * 3 + 1];
      s_cz = bx[f * 3 + 2];
    }
    __syncthreads();
    const float cx = s_cx, cy = s_cy, cz = s_cz;

    float best = -1.0f;
    int bi = 0;
#pragma unroll
    for (int k = 0; k < 8; ++k) {
      const float dx = px[k] - cx, dy = py[k] - cy, dz = pz[k] - cz;
      const float d = dx * dx + dy * dy + dz * dz;
      const float nd = fminf(dist[k], d);
      dist[k] = nd;
      if (nd > best) { best = nd; bi = tid + k * 1024; }
    }
    // wave32 argmax (prefer larger value; tie -> smaller index)
#pragma unroll
    for (int off = 16; off >= 1; off >>= 1) {
      const float ov = __shfl_xor(best, off, 32);
      const int oi = __shfl_xor(bi, off, 32);
      if (ov > best || (ov == best && oi < bi)) { best = ov; bi = oi; }
    }
    if ((tid & 31) == 0) { s_val[tid >> 5] = best; s_idx[tid >> 5] = bi; }
    __syncthreads();
    if (tid < 32) {
      best = s_val[tid];
      bi = s_idx[tid];
#pragma unroll
      for (int off = 16; off >= 1; off >>= 1) {
        const float ov = __shfl_xor(best, off, 32);
        const int oi = __shfl_xor(bi, off, 32);
        if (ov > best || (ov == best && oi < bi)) { best = ov; bi = oi; }
      }
      if (tid == 0) s_far = bi;
    }
    __syncthreads();
  }
}

// ---------------------------------------------------------------------------
// Kernel 2: gather the sampled centroid coordinates into new_xyz.
// ---------------------------------------------------------------------------
__global__ void gather_newxyz_kernel(const float* __restrict__ xyz,
                                     const int* __restrict__ cents,
                                     float* __restrict__ newxyz) {
  const int i = blockIdx.x * blockDim.x + threadIdx.x;  // over B*NPOINT
  if (i >= B_ * NPOINT_) return;
  const int b = i / NPOINT_;
  const int f = cents[i];
  const float* s = xyz + ((size_t)b * N_ + f) * 3;
  newxyz[i * 3 + 0] = s[0];
  newxyz[i * 3 + 1] = s[1];
  newxyz[i * 3 + 2] = s[2];
}

// ---------------------------------------------------------------------------
// Kernel 3: ball query. One wave (32 threads) per tile of 16 centroids.
// Pairwise dot products via V_WMMA_F32_16X16X4_F32:
//   A (16x4) = centroids x (x,y,z,0), B (4x16) = points, D = 16x16 dots.
// dist = |c|^2 + |p|^2 - 2*dot. First-32-in-index-order selection per row
// via wave ballot + prefix popcount (ballot low 16 bits = row j cols 0..15,
// high 16 bits = row j+8 cols 0..15, matching the D VGPR layout).
// ---------------------------------------------------------------------------
__global__ __launch_bounds__(32) void ball_query_wmma_kernel(
    const float* __restrict__ xyz, const float* __restrict__ newxyz,
    int* __restrict__ idxout) {
  const int tile = blockIdx.x;  // B * (NPOINT/16)
  const int b = tile / (NPOINT_ / 16);
  const int cbase = (tile % (NPOINT_ / 16)) * 16;
  const int lane = threadIdx.x;
  const int half = lane >> 4;
  const int l16 = lane & 15;

  const float* bx = xyz + (size_t)b * N_ * 3;
  const float* cx = newxyz + ((size_t)b * NPOINT_ + cbase) * 3;

  // A-matrix: lane 0..15 hold (K0,K1)=(x,y) of row M=l16; lanes 16..31 hold
  // (K2,K3)=(z,0) of the same row.
  const float ax = cx[l16 * 3 + 0];
  const float ay = cx[l16 * 3 + 1];
  const float az = cx[l16 * 3 + 2];
  const float cn2m = ax * ax + ay * ay + az * az;
  v2f A;
  A.x = half ? az : ax;
  A.y = half ? 0.0f : ay;

  // Each lane needs |c|^2 for the 8 rows its D-VGPRs hold: rows 8*half..8*half+7.
  float cn2r[8];
#pragma unroll
  for (int j = 0; j < 8; ++j) cn2r[j] = __shfl(cn2m, j + 8 * half, 32);

  int cnt[8], first[8];
#pragma unroll
  for (int j = 0; j < 8; ++j) { cnt[j] = 0; first[j] = 0; }

  for (int t = 0; t < N_ / 16; ++t) {
    const int p = t * 16 + l16;
    const float qx = bx[p * 3 + 0];
    const float qy = bx[p * 3 + 1];
    const float qz = bx[p * 3 + 2];
    const float pn2 = qx * qx + qy * qy + qz * qz;  // col = l16 for both halves
    // B-matrix 4x16: lanes 0..15 hold rows K0,K1 of col N=l16; lanes 16..31 K2,K3.
    v2f Bm;
    Bm.x = half ? qz : qx;
    Bm.y = half ? 0.0f : qy;

    v8f C = {};
    C = __builtin_amdgcn_wmma_f32_16x16x4_f32(false, A, false, Bm, (short)0, C,
                                              false, false);

    bool alldone = true;
#pragma unroll
    for (int j = 0; j < 8; ++j) {
      const float d2 = cn2r[j] + pn2 - 2.0f * C[j];
      const bool ok = (d2 <= RADIUS2);
      const unsigned bal = (unsigned)__ballot(ok);
      const unsigned m = half ? (bal >> 16) : (bal & 0xFFFFu);  // uniform per half
      if (cnt[j] == 0 && m) first[j] = t * 16 + (__ffs((int)m) - 1);
      if (ok) {
        const int pos = __popc(m & ((1u << l16) - 1u));
        const int slot = cnt[j] + pos;
        if (slot < NSAMPLE_) {
          const int row = j + 8 * half;
          idxout[((size_t)b * NPOINT_ + cbase + row) * NSAMPLE_ + slot] = p;
        }
      }
      cnt[j] += __popc(m);
      alldone = alldone && (cnt[j] >= NSAMPLE_);
    }
    if ((unsigned)__ballot(alldone) == 0xFFFFFFFFu) break;  // uniform exit
  }

  // Pad remaining slots with the first in-radius index (reference semantics).
#pragma unroll
  for (int j = 0; j < 8; ++j) {
    const int row = j + 8 * half;
    for (int s = cnt[j] + l16; s < NSAMPLE_; s += 16) {
      idxout[((size_t)b * NPOINT_ + cbase + row) * NSAMPLE_ + s] = first[j];
    }
  }
}

// ---------------------------------------------------------------------------
// Kernel 4: gather + concat -> new_points (B, NPOINT, NSAMPLE, 3+FEAT).
// One block per centroid; 32x67 contiguous output slab => coalesced stores.
// ---------------------------------------------------------------------------
__global__ __launch_bounds__(256) void group_kernel(
    const float* __restrict__ xyz, const float* __restrict__ points,
    const float* __restrict__ newxyz, const int* __restrict__ idxin,
    float* __restrict__ newpoints) {
  const int ci = blockIdx.x;  // b*NPOINT + i
  const int b = ci / NPOINT_;
  const int tid = threadIdx.x;
  __shared__ int s_idx[NSAMPLE_];
  __shared__ float s_nx[3];
  if (tid < NSAMPLE_) s_idx[tid] = idxin[(size_t)ci * NSAMPLE_ + tid];
  if (tid < 3) s_nx[tid] = newxyz[(size_t)ci * 3 + tid];
  __syncthreads();

  const int CW = 3 + FEAT_;  // 67
  float* outp = newpoints + (size_t)ci * NSAMPLE_ * CW;
  for (int e = tid; e < NSAMPLE_ * CW; e += 256) {
    const int s = e / CW;
    const int c = e % CW;
    const int p = s_idx[s];
    float v;
    if (c < 3)
      v = xyz[((size_t)b * N_ + p) * 3 + c] - s_nx[c];
    else
      v = points[((size_t)b * N_ + p) * FEAT_ + (c - 3)];
    outp[e] = v;
  }
}

// ---------------------------------------------------------------------------
extern "C" void kernel_launch(void* const* d_in, const int* in_sizes, int n_in,
                              void* d_out, int out_size, void* d_ws,
                              size_t ws_size, hipStream_t stream) {
  const float* xyz = (const float*)d_in[0];
  const float* points = (const float*)d_in[1];
  // d_in[2] = npoint (1024), d_in[3] = nsample (32): fixed by setup, hardcoded.

  float* out = (float*)d_out;
  float* newxyz = out;                                   // (16,1024,3)
  float* newpoints = out + (size_t)B_ * NPOINT_ * 3;     // (16,1024,32,67)

  int* cents = (int*)d_ws;                               // B*NPOINT ints
  int* idx = cents + B_ * NPOINT_;                       // B*NPOINT*NSAMPLE ints

  fps_kernel<<<B_, 1024, 0, stream>>>(xyz, cents);
  gather_newxyz_kernel<<<(B_ * NPOINT_ + 255) / 256, 256, 0, stream>>>(
      xyz, cents, newxyz);
  ball_query_wmma_kernel<<<B_ * (NPOINT_ / 16), 32, 0, stream>>>(xyz, newxyz,
                                                                 idx);
  group_kernel<<<B_ * NPOINT_, 256, 0, stream>>>(xyz, points, newxyz, idx,
                                                 newpoints);
}